// EmbeddingIn_24532853194881
// MI455X (gfx1250) — compile-verified
//
#include <hip/hip_runtime.h>
#include <hip/hip_bf16.h>
#include <stdint.h>

// Problem constants (from the reference)
#define N_TOKEN   35
#define EMB_DIM   512
#define P_LEN     130          // MAX_LEN + 2
#define START_ID  32
#define END_ID    33
#define NULL_ID   34

#ifndef __has_builtin
#define __has_builtin(x) 0
#endif

// ---- CDNA5 async LDS<->global path (probe-guarded) -------------------------
#if __has_builtin(__builtin_amdgcn_global_load_async_to_lds_b128) && \
    __has_builtin(__builtin_amdgcn_global_store_async_from_lds_b128)
#define HAVE_ASYNC 1
#else
#define HAVE_ASYNC 0
#endif

#if __has_builtin(__builtin_amdgcn_s_wait_asynccnt)
#define WAIT_ASYNC() __builtin_amdgcn_s_wait_asynccnt(0)
#else
#define WAIT_ASYNC() asm volatile("s_wait_asynccnt 0" ::: "memory")
#endif

typedef int   v4i __attribute__((vector_size(16)));
typedef float v4f __attribute__((vector_size(16)));

#if HAVE_ASYNC
#define AS1 __attribute__((address_space(1)))
#define AS3 __attribute__((address_space(3)))
typedef AS1 v4i* g4p;   // global int4*
typedef AS3 v4i* l4p;   // LDS int4*

// gfx12+ CPol: TH[2:0] in bits [2:0]; store TH=1 -> NT (non-temporal)
#define CPOL_RT 0
#define CPOL_NT 1

__device__ __forceinline__ void async_g2l_b128(const float* g, float* l) {
  // global -> LDS, 16 bytes per lane, tracked on ASYNCcnt (RT: emb stays hot)
  __builtin_amdgcn_global_load_async_to_lds_b128((g4p)(v4i*)g, (l4p)(v4i*)l, 0, CPOL_RT);
}
__device__ __forceinline__ void async_l2g_b128(float* g, const float* l) {
  // LDS -> global, 16 bytes per lane, NT store: 1.09GB write-once stream
  __builtin_amdgcn_global_store_async_from_lds_b128((g4p)(v4i*)g, (l4p)(v4i*)l, 0, CPOL_NT);
}
#endif

// ---------------------------------------------------------------------------
// Kernel 1: exclusive prefix sum of src_len (replaces cumsum). One block.
// ---------------------------------------------------------------------------
__global__ __launch_bounds__(1024)
void EmbeddingIn_scan_kernel(const int* __restrict__ len,
                             int* __restrict__ starts, int n) {
  __shared__ int tmp[1024];
  const int t = threadIdx.x;
  const int base = t * 4;
  const int a0 = (base + 0 < n) ? len[base + 0] : 0;
  const int a1 = (base + 1 < n) ? len[base + 1] : 0;
  const int a2 = (base + 2 < n) ? len[base + 2] : 0;
  const int a3 = (base + 3 < n) ? len[base + 3] : 0;
  const int s0 = a0, s1 = s0 + a1, s2 = s1 + a2, s3 = s2 + a3;
  tmp[t] = s3;
  __syncthreads();
  for (int off = 1; off < 1024; off <<= 1) {
    const int u = (t >= off) ? tmp[t - off] : 0;
    __syncthreads();
    tmp[t] += u;
    __syncthreads();
  }
  const int excl = tmp[t] - s3;   // exclusive prefix of this thread's chunk
  if (base + 0 < n) starts[base + 0] = excl;
  if (base + 1 < n) starts[base + 1] = excl + s0;
  if (base + 2 < n) starts[base + 2] = excl + s1;
  if (base + 3 < n) starts[base + 3] = excl + s2;
}

// ---------------------------------------------------------------------------
// Kernel 2: one 256-thread block per batch row.
//   - async-stage emb (35x512 f32 = 71.7KB) into LDS
//   - compute the 130 token ids for this row (overlapped with the async copy)
//   - stream 130 rows x 2KB out of LDS with NT async b128 stores
// ---------------------------------------------------------------------------
__global__ __launch_bounds__(256)
void EmbeddingIn_gather_kernel(const int* __restrict__ src,
                               const int* __restrict__ len,
                               const float* __restrict__ emb,
                               const int* __restrict__ starts,
                               float* __restrict__ out, int total) {
  extern __shared__ float smem[];
  float* semb = smem;                                   // N_TOKEN*EMB_DIM floats
  int*   sids = (int*)(smem + N_TOKEN * EMB_DIM);       // P_LEN ids

  const int tid = threadIdx.x;
  const int b   = blockIdx.x;

  // ---- stage embedding table into LDS (4480 x 16B chunks) ----
  const int NV4 = (N_TOKEN * EMB_DIM) / 4;
#if HAVE_ASYNC
  for (int i = tid; i < NV4; i += 256)
    async_g2l_b128(emb + 4 * i, semb + 4 * i);
#else
  for (int i = tid; i < NV4; i += 256)
    ((v4f*)semb)[i] = ((const v4f*)emb)[i];
#endif

  // ---- per-position token id (reference's nested where) ----
  if (tid < P_LEN) {
    const int p = tid;
    const int l = len[b];
    int id;
    if (p == 0) {
      id = START_ID;
    } else if (p <= l) {
      int g = starts[b] + p - 1;
      g = (g < total - 1) ? g : (total - 1);
      id = src[g];
    } else if (p == l + 1) {
      id = END_ID;
    } else {
      id = NULL_ID;
    }
    sids[p] = id;
  }

#if HAVE_ASYNC
  WAIT_ASYNC();          // LDS copy of emb complete for this wave
#endif
  __syncthreads();        // all waves: emb + sids visible

  // ---- stream output rows: wave w handles rows w, w+8, ... ----
  const int wave = tid >> 5;
  const int lane = tid & 31;
  float* outb = out + (size_t)b * (P_LEN * EMB_DIM);
  for (int row = wave; row < P_LEN; row += 8) {
    const int id = sids[row];
    const float* srow = semb + id * EMB_DIM;
    float*       drow = outb + row * EMB_DIM;
#if HAVE_ASYNC
#pragma unroll
    for (int j = 0; j < 4; ++j)   // 32 lanes x 4 x 16B = 2KB row
      async_l2g_b128(drow + 4 * lane + 128 * j, srow + 4 * lane + 128 * j);
#else
#pragma unroll
    for (int j = 0; j < 4; ++j) {
      const v4f v = ((const v4f*)srow)[lane + 32 * j];
      __builtin_nontemporal_store(v, (v4f*)drow + lane + 32 * j);
    }
#endif
  }
#if HAVE_ASYNC
  WAIT_ASYNC();          // drain async stores (S_ENDPGM also waits idle)
#endif
}

// ---------------------------------------------------------------------------
extern "C" void kernel_launch(void* const* d_in, const int* in_sizes, int n_in,
                              void* d_out, int out_size, void* d_ws, size_t ws_size,
                              hipStream_t stream) {
  const int*   src     = (const int*)d_in[0];
  const int*   src_len = (const int*)d_in[1];
  const float* emb     = (const float*)d_in[2];
  const int total = in_sizes[0];
  const int batch = in_sizes[1];

  int* starts = (int*)d_ws;   // batch int32 exclusive prefix sums

  EmbeddingIn_scan_kernel<<<1, 1024, 0, stream>>>(src_len, starts, batch);

  const size_t shmem = (size_t)(N_TOKEN * EMB_DIM) * sizeof(float) +
                       (size_t)P_LEN * sizeof(int);
  EmbeddingIn_gather_kernel<<<dim3(batch), dim3(256), shmem, stream>>>(
      src, src_len, emb, starts, (float*)d_out, total);
}